// TM_73727408603569
// MI455X (gfx1250) — compile-verified
//
#include <hip/hip_runtime.h>
#include <math.h>

// B-spline constants (match reference): ORDER=3, N_COEF=32,
// range [-15,15], span_factor 0.1 -> lo=-18, hi=18, dist=36/29.
#define BSP_ORDER 3
#define BSP_NCOEF 32
#define BSP_NSPAN 29                 // reachable spans: s in [3,31] -> b = s-3 in [0,28]
#define BSP_LO    (-18.0f)
#define BSP_DIST  (36.0f / 29.0f)
#define BSP_INV1  (29.0f / 36.0f)    // 1/dist
#define BSP_INV2  (29.0f / 72.0f)    // 1/(2*dist)
#define BSP_INV3  (29.0f / 108.0f)   // 1/(3*dist)
#define HALF_LOG_2PI 0.91893853320467274f

__device__ __forceinline__ float softplus_f(float x) {
    // stable: max(x,0) + log1p(exp(-|x|))
    return fmaxf(x, 0.0f) + log1pf(__expf(-fabsf(x)));
}

// ---- Kernel 1: tiny preprocessing (one wave).
// Emits a span-packed table: for each span b in [0,29):
//   tbl[8b+0..3] = coefs[b..b+3]   (cubic de Boor window)
//   tbl[8b+4..6] = dcoefs[b..b+2]  (quadratic de Boor window)
//   tbl[8b+7]    = 0               (pad to 32B so float4 loads stay aligned)
__global__ void TM_bspline_prep(const float* __restrict__ params,
                                float* __restrict__ tbl) {
    if (threadIdx.x == 0) {
        float c[BSP_NCOEF];
        float dc[BSP_NCOEF - 1];
        c[0] = params[0];
        float acc = params[0];
        for (int k = 1; k < BSP_NCOEF; ++k) {
            acc += softplus_f(params[k]);
            c[k] = acc;
        }
        // dcoefs[i] = 3*(c[i+1]-c[i]) / (kn(i+4)-kn(i+1)); kn(i) = lo + (i-3)*dist
        for (int i = 0; i < BSP_NCOEF - 1; ++i) {
            float kr = fmaf((float)(i + 4 - BSP_ORDER), BSP_DIST, BSP_LO);
            float kl = fmaf((float)(i + 1 - BSP_ORDER), BSP_DIST, BSP_LO);
            dc[i] = 3.0f * (c[i + 1] - c[i]) / (kr - kl);
        }
        for (int b = 0; b < BSP_NSPAN; ++b) {
            tbl[8 * b + 0] = c[b + 0];
            tbl[8 * b + 1] = c[b + 1];
            tbl[8 * b + 2] = c[b + 2];
            tbl[8 * b + 3] = c[b + 3];
            tbl[8 * b + 4] = dc[b + 0];
            tbl[8 * b + 5] = dc[b + 1];
            tbl[8 * b + 6] = dc[b + 2];
            tbl[8 * b + 7] = 0.0f;
        }
    }
}

__device__ __forceinline__ float lerp_f(float a, float b, float t) {
    return fmaf(t, b - a, a);
}

__device__ __forceinline__ float eval_one(float yv, const float* __restrict__ sp) {
    float x = fminf(fmaxf(yv, -15.0f), 15.0f);
    float t = (x - BSP_LO) * BSP_INV1;           // t in [2.41, 26.59] -> trunc == floor
    int b = min(max((int)t, 0), BSP_NSPAN - 1);  // b = span - ORDER, clipped like reference

    // u = x - kn(span); all de Boor alphas are FMAs of u against fixed reciprocals
    float u  = x - fmaf((float)b, BSP_DIST, BSP_LO);
    float u1 = u + BSP_DIST;
    float u2 = u + 2.0f * BSP_DIST;

    float a13 = u  * BSP_INV3;   // cubic r=1
    float a12 = u1 * BSP_INV3;
    float a11 = u2 * BSP_INV3;
    float a23 = u  * BSP_INV2;   // cubic r=2 == quad r=1
    float a22 = u1 * BSP_INV2;
    float a33 = u  * BSP_INV1;   // cubic r=3 == quad r=2

    // two aligned ds_load_b128: {c0..c3} and {dc0..dc2,pad}
    float4 cv = *(const float4*)(sp + 8 * b);
    float4 dv = *(const float4*)(sp + 8 * b + 4);

    // cubic de Boor (value)
    float d1 = cv.y, d2 = cv.z, d3 = cv.w;
    d3 = lerp_f(d2, d3, a13);
    d2 = lerp_f(d1, d2, a12);
    d1 = lerp_f(cv.x, d1, a11);
    d3 = lerp_f(d2, d3, a23);
    d2 = lerp_f(d1, d2, a22);
    d3 = lerp_f(d2, d3, a33);

    // quadratic de Boor (derivative), reuses a23/a22/a33
    float e1 = dv.y, e2 = dv.z;
    e2 = lerp_f(e1, e2, a23);
    e1 = lerp_f(dv.x, e1, a22);
    e2 = lerp_f(e1, e2, a33);

    // -0.5*z^2 - 0.5*log(2pi) + log(der); v_log_f32 is a TRANS op, co-executes with VALU
    return fmaf(-0.5f * d3, d3, -HALF_LOG_2PI) + __logf(e2);
}

// ---- Kernel 2: streaming float4 (global_load/store_b128) + packed LDS table
__global__ void __launch_bounds__(256)
TM_bspline_logprob(const float* __restrict__ y,
                   const float* __restrict__ tbl,
                   float* __restrict__ out,
                   int n4, int n) {
    __shared__ float4 spk[2 * BSP_NSPAN];        // 928 B
    int tid = threadIdx.x;
    if (tid < 2 * BSP_NSPAN) spk[tid] = ((const float4*)tbl)[tid];
    __syncthreads();
    const float* sp = (const float*)spk;

    const float4* __restrict__ y4   = (const float4*)y;
    float4* __restrict__       out4 = (float4*)out;

    int idx    = blockIdx.x * blockDim.x + threadIdx.x;
    int stride = gridDim.x * blockDim.x;
    for (int i = idx; i < n4; i += stride) {
        // gfx1250 global_prefetch_b8 for the next strided chunk (speculative)
        __builtin_prefetch(&y4[i + stride], 0, 0);
        float4 v = y4[i];
        float4 r;
        r.x = eval_one(v.x, sp);
        r.y = eval_one(v.y, sp);
        r.z = eval_one(v.z, sp);
        r.w = eval_one(v.w, sp);
        out4[i] = r;
    }

    // scalar tail (n % 4 elements)
    int tail = n - n4 * 4;
    if (blockIdx.x == 0 && tid < tail) {
        int j = n4 * 4 + tid;
        out[j] = eval_one(y[j], sp);
    }
}

extern "C" void kernel_launch(void* const* d_in, const int* in_sizes, int n_in,
                              void* d_out, int out_size, void* d_ws, size_t ws_size,
                              hipStream_t stream) {
    const float* params = (const float*)d_in[0];   // [32]
    const float* y      = (const float*)d_in[1];   // [20M]
    float* out          = (float*)d_out;
    float* tbl          = (float*)d_ws;            // 232 floats (928 B) of scratch
    int n  = in_sizes[1];
    int n4 = n >> 2;

    TM_bspline_prep<<<1, 32, 0, stream>>>(params, tbl);

    const int threads = 256;                       // 8 wave32 waves / block
    int blocks = (n4 + threads - 1) / threads;
    if (blocks > 8192) blocks = 8192;              // >=2 float4 iters/thread -> prefetch useful
    if (blocks < 1) blocks = 1;
    TM_bspline_logprob<<<blocks, threads, 0, stream>>>(y, tbl, out, n4, n);
}